// CrossFeature_28673201668370
// MI455X (gfx1250) — compile-verified
//
#include <hip/hip_runtime.h>
#include <hip/hip_bf16.h>
#include <stdint.h>

// Problem constants (match the reference)
#define F_DIM   300      // features
#define C_DIM   4        // latent dim
#define B_DIM   8192     // batch
#define GP      304      // Gram matrix padded leading dim (19*16)
#define NTILE   19       // 16x16 tiles per Gram dimension
#define P_PAIRS 44850    // F*(F-1)/2
#define P_PAD   44864    // P rounded up to multiple of 16 (for ws alignment)

// Workspace layout (in floats):
//   [0,                 92416)  : padded Gram matrix g[304][304]
//   [92416,     92416+44864)    : w[p]   (pairwise dot products)
//   [137280,   137280+44864)    : idx[p] (packed (i<<16)|j, as uint32)
// Total: 182144 floats = 728,576 bytes.
#define WS_GRAM_OFF 0
#define WS_W_OFF    92416
#define WS_IDX_OFF  137280

typedef float    v2f  __attribute__((ext_vector_type(2)));
typedef float    v8f  __attribute__((ext_vector_type(8)));
typedef float    f2v  __attribute__((ext_vector_type(2)));
typedef uint32_t u2v  __attribute__((ext_vector_type(2)));

// ---------------------------------------------------------------------------
// Kernel 1: Gram matrix g = v * v^T via V_WMMA_F32_16X16X4_F32.
// One wave (32 threads) per 16x16 output tile; 19x19 = 361 tiles.
// A-operand layout (ISA 7.12.2, 32-bit A 16x4): lanes 0-15 -> M=lane, K={0,1};
// lanes 16-31 -> M=lane-16, K={2,3}. B (4x16, B = v^T so B[k,n] = v[n,k]) is
// loaded with the mirrored layout over N. C/D: 8 VGPRs, lane<16 -> M=r,
// lane>=16 -> M=r+8, N = lane&15.
// ---------------------------------------------------------------------------
__global__ __launch_bounds__(32)
void gram_wmma_kernel(const float* __restrict__ v, float* __restrict__ gram) {
    const int tile = blockIdx.x;
    const int m0 = (tile / NTILE) * 16;
    const int n0 = (tile % NTILE) * 16;
    const int lane = threadIdx.x;
    const int half = lane >> 4;        // 0: K={0,1}   1: K={2,3}
    const int r    = lane & 15;
    const int k0   = half * 2;

    const int am = m0 + r;             // A row (M) for this lane
    const int bn = n0 + r;             // B col (N) for this lane

    v2f a, b;
    a.x = (am < F_DIM) ? v[am * C_DIM + k0]     : 0.0f;
    a.y = (am < F_DIM) ? v[am * C_DIM + k0 + 1] : 0.0f;
    b.x = (bn < F_DIM) ? v[bn * C_DIM + k0]     : 0.0f;
    b.y = (bn < F_DIM) ? v[bn * C_DIM + k0 + 1] : 0.0f;

    v8f c = {0.f, 0.f, 0.f, 0.f, 0.f, 0.f, 0.f, 0.f};
    // 8 args: (neg_a, A, neg_b, B, c_mod, C, reuse_a, reuse_b)
    c = __builtin_amdgcn_wmma_f32_16x16x4_f32(
        false, a, false, b, (short)0, c, false, false);

    // Scatter D tile into the padded Gram buffer (always in-bounds: GP=304).
    const int col = n0 + r;
#pragma unroll
    for (int rr = 0; rr < 8; ++rr) {
        const int row = m0 + rr + half * 8;
        gram[row * GP + col] = c[rr];
    }
}

// ---------------------------------------------------------------------------
// Kernel 2: build the pair tables in row-major upper-triangular order
// (identical order to jnp.triu_indices / the nested python loop).
// Block i handles row i: pairs p = base(i) + (j - i - 1), j in (i, F).
// ---------------------------------------------------------------------------
__global__ __launch_bounds__(128)
void pairs_kernel(const float* __restrict__ gram,
                  float* __restrict__ w, uint32_t* __restrict__ idx) {
    const int i = blockIdx.x;                         // 0 .. F-2
    const int base = (i * (2 * F_DIM - i - 1)) >> 1;  // offset of row i's pairs
    for (int j = i + 1 + threadIdx.x; j < F_DIM; j += blockDim.x) {
        const int p = base + (j - i - 1);
        w[p]   = gram[i * GP + j];
        idx[p] = ((uint32_t)i << 16) | (uint32_t)j;
    }
}

// ---------------------------------------------------------------------------
// Kernel 3: streaming epilogue. One block per batch row; x-row staged in LDS;
// w/idx stay hot in L2 (358 KB << 192 MB); output (1.47 GB, touched once)
// streamed with non-temporal b64 stores. P is even -> exact float2 tiling;
// row byte stride 179400 is 8B-aligned -> float2 accesses always aligned.
// ---------------------------------------------------------------------------
__global__ __launch_bounds__(512)
void cross_main_kernel(const float* __restrict__ x,
                       const float* __restrict__ w,
                       const uint32_t* __restrict__ idx,
                       float* __restrict__ out) {
    __shared__ float sx[F_DIM];
    const int b = blockIdx.x;
    const float* xr = x + (size_t)b * F_DIM;
    for (int t = threadIdx.x; t < F_DIM; t += blockDim.x) sx[t] = xr[t];
    __syncthreads();

    float* orow = out + (size_t)b * P_PAIRS;
    const int NV2 = P_PAIRS / 2;                      // 22425 float2 chunks
    for (int q = threadIdx.x; q < NV2; q += blockDim.x) {
        const int p = q * 2;
        const u2v id2 = *reinterpret_cast<const u2v*>(idx + p);
        const f2v w2  = *reinterpret_cast<const f2v*>(w + p);
        f2v r;
        r.x = sx[id2.x >> 16] * sx[id2.x & 0xFFFFu] * w2.x;
        r.y = sx[id2.y >> 16] * sx[id2.y & 0xFFFFu] * w2.y;
        __builtin_nontemporal_store(r, reinterpret_cast<f2v*>(orow + p));
    }
}

// ---------------------------------------------------------------------------
extern "C" void kernel_launch(void* const* d_in, const int* in_sizes, int n_in,
                              void* d_out, int out_size, void* d_ws, size_t ws_size,
                              hipStream_t stream) {
    const float* x = (const float*)d_in[0];   // [B, F] f32
    const float* v = (const float*)d_in[1];   // [F, C] f32
    float* out = (float*)d_out;               // [B, P] f32

    float*    ws_gram = (float*)d_ws + WS_GRAM_OFF;
    float*    ws_w    = (float*)d_ws + WS_W_OFF;
    uint32_t* ws_idx  = (uint32_t*)((float*)d_ws + WS_IDX_OFF);

    // 1) Gram matrix via f32 WMMA (361 waves).
    gram_wmma_kernel<<<NTILE * NTILE, 32, 0, stream>>>(v, ws_gram);
    // 2) Pair tables (w[p], packed (i,j)).
    pairs_kernel<<<F_DIM - 1, 128, 0, stream>>>(ws_gram, ws_w, ws_idx);
    // 3) Streaming NT-store epilogue: one block per batch row.
    cross_main_kernel<<<B_DIM, 512, 0, stream>>>(x, ws_w, ws_idx, out);
}